// FeatureEmbedding_41068477284866
// MI455X (gfx1250) — compile-verified
//
#include <hip/hip_runtime.h>

typedef __attribute__((ext_vector_type(2))) float v2f;
typedef __attribute__((ext_vector_type(4))) float v4f;
typedef __attribute__((ext_vector_type(8))) float v8f;

#define INPUT_DIM 128
#define D_MODEL   128
#define BATCH     8192
#define LN_EPS    1e-5f

// ---------------------------------------------------------------------------
// Kernel 1: per-feature moments of W and b via V_WMMA_F32_16X16X4_F32.
// One wave handles 16 features. Row sums = A x Ones (B = all-ones, so B's
// VGPR layout is irrelevant). A-layout (16x4 f32, ISA 7.12.2): lane<16 holds
// [M=lane, K=k0..k0+1], lane>=16 holds [M=lane-16, K=k0+2..k0+3].
// D-layout: VGPR r holds M=r (lanes 0-15) / M=8+r (lanes 16-31); all 16
// columns carry the same row-sum, so lanes 0 and 16 extract everything.
// ---------------------------------------------------------------------------
__global__ __launch_bounds__(32) void feat_stats_wmma_kernel(
    const float* __restrict__ W, const float* __restrict__ b,
    float* __restrict__ statA, float* __restrict__ statB, float* __restrict__ statC,
    float* __restrict__ mwArr, float* __restrict__ mbArr)
{
    const int lane = threadIdx.x;
    const int half = lane >> 4;      // 0: K=0..1, 1: K=2..3
    const int m    = lane & 15;      // row within 16-feature group
    const int f0   = blockIdx.x * 16;
    const int row  = f0 + m;

    const float* wp = W + (size_t)row * D_MODEL;
    const float* bp = b + (size_t)row * D_MODEL;

    v8f accW = {}, accB = {}, accW2 = {}, accB2 = {}, accWB = {};
    v2f ones; ones.x = 1.0f; ones.y = 1.0f;

    for (int k0 = 0; k0 < D_MODEL; k0 += 4) {
        v2f aw = *(const v2f*)(wp + k0 + 2 * half);
        v2f ab = *(const v2f*)(bp + k0 + 2 * half);
        v2f aw2 = aw * aw;
        v2f ab2 = ab * ab;
        v2f awb = aw * ab;
        accW  = __builtin_amdgcn_wmma_f32_16x16x4_f32(false, aw,  false, ones, (short)0, accW,  false, false);
        accB  = __builtin_amdgcn_wmma_f32_16x16x4_f32(false, ab,  false, ones, (short)0, accB,  false, false);
        accW2 = __builtin_amdgcn_wmma_f32_16x16x4_f32(false, aw2, false, ones, (short)0, accW2, false, false);
        accB2 = __builtin_amdgcn_wmma_f32_16x16x4_f32(false, ab2, false, ones, (short)0, accB2, false, false);
        accWB = __builtin_amdgcn_wmma_f32_16x16x4_f32(false, awb, false, ones, (short)0, accWB, false, false);
    }

    if (m == 0) {  // lanes 0 (features f0..f0+7) and 16 (f0+8..f0+15)
        const float inv = 1.0f / (float)D_MODEL;
        #pragma unroll
        for (int r = 0; r < 8; ++r) {
            const int f  = f0 + half * 8 + r;
            const float mw  = accW[r]  * inv;
            const float mb  = accB[r]  * inv;
            const float Sww = accW2[r] * inv - mw * mw;
            const float Swb = accWB[r] * inv - mw * mb;
            const float Sbb = accB2[r] * inv - mb * mb;
            statA[f] = Sww;               // var = A*x^2 + B*x + C  (eps folded in C)
            statB[f] = 2.0f * Swb;
            statC[f] = Sbb + LN_EPS;
            mwArr[f] = mw;
            mbArr[f] = mb;
        }
    }
}

// ---------------------------------------------------------------------------
// Kernel 2: centered, gamma-scaled tables  wg=(W-mw)*gamma, bg=(b-mb)*gamma.
// ---------------------------------------------------------------------------
__global__ __launch_bounds__(128) void prep_tables_kernel(
    const float* __restrict__ W, const float* __restrict__ b,
    const float* __restrict__ gamma,
    const float* __restrict__ mwArr, const float* __restrict__ mbArr,
    float* __restrict__ wg, float* __restrict__ bg)
{
    const int f = blockIdx.x;
    const int d = threadIdx.x;
    const float g  = gamma[d];
    const size_t i = (size_t)f * D_MODEL + d;
    wg[i] = (W[i] - mwArr[f]) * g;
    bg[i] = (b[i] - mbArr[f]) * g;
}

// ---------------------------------------------------------------------------
// Kernel 3: streaming producer. One wave owns 32 consecutive rows (n) of one
// feature f; lane l covers d = 4l..4l+3. Per row: closed-form rsqrt + 2 FMAs
// per element, one coalesced 512B row store (NT: output >> 192MB L2).
// ---------------------------------------------------------------------------
__global__ __launch_bounds__(256) void emb_ln_stream_kernel(
    const float* __restrict__ x,
    const float* __restrict__ wg, const float* __restrict__ bg,
    const float* __restrict__ beta,
    const float* __restrict__ statA, const float* __restrict__ statB,
    const float* __restrict__ statC,
    float* __restrict__ out)
{
    const int lane = threadIdx.x;   // 0..31
    const int wave = threadIdx.y;   // 0..7
    const int f    = blockIdx.y;
    const int d0   = lane * 4;

    const v4f wg4 = *(const v4f*)(wg + (size_t)f * D_MODEL + d0);
    const v4f bg4 = *(const v4f*)(bg + (size_t)f * D_MODEL + d0);
    const v4f be4 = *(const v4f*)(beta + d0);
    const float A = statA[f];
    const float B = statB[f];
    const float C = statC[f];

    const int nbase = blockIdx.x * 256 + wave * 32;
    // lane l pre-loads x for row (nbase + l); broadcast via shuffle below.
    const float xv = x[(size_t)(nbase + lane) * INPUT_DIM + f];

    float* orow = out + ((size_t)f * BATCH + nbase) * D_MODEL + d0;

    #pragma unroll 4
    for (int i = 0; i < 32; ++i) {
        const float xi  = __shfl(xv, i, 32);
        const float rsq = rsqrtf(fmaf(xi, fmaf(xi, A, B), C));
        v4f o;
        o.x = fmaf(rsq, fmaf(xi, wg4.x, bg4.x), be4.x);
        o.y = fmaf(rsq, fmaf(xi, wg4.y, bg4.y), be4.y);
        o.z = fmaf(rsq, fmaf(xi, wg4.z, bg4.z), be4.z);
        o.w = fmaf(rsq, fmaf(xi, wg4.w, bg4.w), be4.w);
        __builtin_nontemporal_store(o, (v4f*)(orow + (size_t)i * D_MODEL));
    }
}

// ---------------------------------------------------------------------------
extern "C" void kernel_launch(void* const* d_in, const int* in_sizes, int n_in,
                              void* d_out, int out_size, void* d_ws, size_t ws_size,
                              hipStream_t stream) {
    (void)in_sizes; (void)n_in; (void)out_size; (void)ws_size;

    const float* x     = (const float*)d_in[0];
    const float* W     = (const float*)d_in[1];
    const float* b     = (const float*)d_in[2];
    const float* gamma = (const float*)d_in[3];
    const float* beta  = (const float*)d_in[4];
    float* out = (float*)d_out;

    // workspace layout (floats): wg[16384] | bg[16384] | A[128] B[128] C[128] mw[128] mb[128]
    float* ws    = (float*)d_ws;
    float* wg    = ws;
    float* bg    = ws + 16384;
    float* statA = ws + 32768;
    float* statB = statA + 128;
    float* statC = statB + 128;
    float* mwA   = statC + 128;
    float* mbA   = mwA + 128;

    feat_stats_wmma_kernel<<<dim3(8), dim3(32), 0, stream>>>(W, b, statA, statB, statC, mwA, mbA);
    prep_tables_kernel<<<dim3(128), dim3(128), 0, stream>>>(W, b, gamma, mwA, mbA, wg, bg);

    dim3 blk(32, 8);
    dim3 grd(BATCH / 256, INPUT_DIM);
    emb_ln_stream_kernel<<<grd, blk, 0, stream>>>(x, wg, bg, beta, statA, statB, statC, out);
}